// TopKSequenceRouter_74929999446933
// MI455X (gfx1250) — compile-verified
//
#include <hip/hip_runtime.h>
#include <hip/hip_bf16.h>

typedef __attribute__((ext_vector_type(2))) float v2f;
typedef __attribute__((ext_vector_type(8))) float v8f;

#define NUM_EXPERTS 64
#define TOPK 8
#define HDIM 2048
#define SEQ 8
#define ROWS_PER_BLOCK 16

// ---------------------------------------------------------------------------
// Main router kernel: 1 block = 16 rows of B, 128 threads = 4 waves.
// Wave w computes the 16x16 logits tile for experts [16w, 16w+16) with
// V_WMMA_F32_16X16X4_F32, then the block does softmax + top-8 in LDS.
// ---------------------------------------------------------------------------
__global__ __launch_bounds__(128) void router_kernel(
    const float* __restrict__ hidden,   // [B, SEQ, HDIM], only seq 0 used
    const float* __restrict__ gate_w,   // [E, HDIM]
    int*   __restrict__ out_idx,        // [B, TOPK]
    float* __restrict__ out_w,          // [B, TOPK]
    float* __restrict__ ws_importance,  // [E] raw prob sums over B
    float* __restrict__ ws_load)        // [E] raw selection counts
{
    __shared__ float probsS[ROWS_PER_BLOCK][NUM_EXPERTS + 1];
    __shared__ float loadS[NUM_EXPERTS];

    const int tid  = threadIdx.x;
    const int lane = tid & 31;
    const int wave = tid >> 5;                 // 0..3 -> expert column tile
    const int rowBase    = blockIdx.x * ROWS_PER_BLOCK;
    const int expertBase = wave * 16;

    if (tid < NUM_EXPERTS) loadS[tid] = 0.0f;

    // ---- WMMA GEMM: D[16x16] = A[16x2048] * B[2048x16], K-step 4 ----------
    // ISA 32-bit A 16x4 layout: lanes 0-15 hold K={k,k+1}, lanes 16-31 hold
    // K={k+2,k+3}; M (rows) = lane&15.  B (4x16) is the mirror: N = lane&15.
    const int mRow  = lane & 15;
    const int kHalf = (lane >> 4) << 1;        // 0 or 2

    const float* aPtr = hidden + (size_t)(rowBase + mRow) * (SEQ * HDIM) + kHalf;
    const float* bPtr = gate_w + (size_t)(expertBase + mRow) * HDIM + kHalf;

    v8f c = {};
    #pragma unroll 8
    for (int k = 0; k < HDIM; k += 4) {
        v2f a = *(const v2f*)(aPtr);
        v2f b = *(const v2f*)(bPtr);
        c = __builtin_amdgcn_wmma_f32_16x16x4_f32(
                /*neg_a=*/false, a, /*neg_b=*/false, b,
                /*c_mod=*/(short)0, c, /*reuse_a=*/false, /*reuse_b=*/false);
        aPtr += 4;
        bPtr += 4;
    }

    // C/D layout: VGPR r of lane L holds (M = r + 8*(L>=16), N = L&15).
    {
        const int nCol = lane & 15;
        const int mOff = (lane >> 4) * 8;
        #pragma unroll
        for (int r = 0; r < 8; ++r)
            probsS[mOff + r][expertBase + nCol] = c[r];
    }
    __syncthreads();

    // ---- Softmax + top-8 per row (16 threads, one row each) ----------------
    if (tid < ROWS_PER_BLOCK) {
        const int row = tid;

        float m = -3.4e38f;
        #pragma unroll 8
        for (int e = 0; e < NUM_EXPERTS; ++e) m = fmaxf(m, probsS[row][e]);

        float s = 0.0f;
        #pragma unroll 8
        for (int e = 0; e < NUM_EXPERTS; ++e) {
            float p = __expf(probsS[row][e] - m);
            probsS[row][e] = p;
            s += p;
        }
        const float invS = 1.0f / s;
        #pragma unroll 8
        for (int e = 0; e < NUM_EXPERTS; ++e) probsS[row][e] *= invS;

        int   idxs[TOPK];
        float vals[TOPK];
        unsigned long long used = 0ull;
        float tsum = 0.0f;
        for (int j = 0; j < TOPK; ++j) {
            int best = 0; float bv = -1.0f;
            for (int e = 0; e < NUM_EXPERTS; ++e) {
                float p = probsS[row][e];
                if (!((used >> e) & 1ull) && p > bv) { bv = p; best = e; }
            }
            used |= 1ull << best;
            idxs[j] = best; vals[j] = bv; tsum += bv;
            atomicAdd(&loadS[best], 1.0f);
        }
        const float invT = 1.0f / fmaxf(tsum, 1e-8f);
        const int gRow = rowBase + row;
        #pragma unroll
        for (int j = 0; j < TOPK; ++j) {
            out_idx[gRow * TOPK + j] = idxs[j];
            out_w[gRow * TOPK + j]   = vals[j] * invT;
        }
    }
    __syncthreads();

    // ---- Per-block importance / load partials -> global accumulators ------
    if (tid < NUM_EXPERTS) {
        float s = 0.0f;
        #pragma unroll
        for (int r = 0; r < ROWS_PER_BLOCK; ++r) s += probsS[r][tid];
        atomicAdd(&ws_importance[tid], s);
        atomicAdd(&ws_load[tid], loadS[tid]);
    }
}

// ---------------------------------------------------------------------------
// Fold accumulators into the scalar aux loss.
// aux = sum_e  E * (imp[e]/B) * (load[e]/(B*K))
// ---------------------------------------------------------------------------
__global__ void aux_kernel(const float* __restrict__ imp,
                           const float* __restrict__ ld,
                           float* __restrict__ out_scalar, int B)
{
    __shared__ float red[NUM_EXPERTS];
    const int e = threadIdx.x;
    const float importance = imp[e] / (float)B;
    const float load       = ld[e]  / (float)(B * TOPK);
    red[e] = (float)NUM_EXPERTS * importance * load;
    __syncthreads();
    for (int s = NUM_EXPERTS / 2; s > 0; s >>= 1) {
        if (e < s) red[e] += red[e + s];
        __syncthreads();
    }
    if (e == 0) *out_scalar = red[0];
}

__global__ void zero_kernel(float* __restrict__ ws) {
    ws[threadIdx.x] = 0.0f;
}

extern "C" void kernel_launch(void* const* d_in, const int* in_sizes, int n_in,
                              void* d_out, int out_size, void* d_ws, size_t ws_size,
                              hipStream_t stream) {
    const float* hidden = (const float*)d_in[0];   // [B, 8, 2048] f32
    const float* gate_w = (const float*)d_in[1];   // [64, 2048] f32
    const int B = in_sizes[0] / (SEQ * HDIM);

    // Output layout (return order): topk_idx [B,8] i32, topk_weights [B,8] f32,
    // aux_loss scalar f32 — all 4-byte, concatenated flat.
    int*   out_idx = (int*)d_out;
    float* out_w   = (float*)d_out + (size_t)B * TOPK;
    float* out_aux = (float*)d_out + (size_t)2 * B * TOPK;

    float* ws_imp  = (float*)d_ws;                 // [64]
    float* ws_load = ws_imp + NUM_EXPERTS;         // [64]

    zero_kernel<<<1, 2 * NUM_EXPERTS, 0, stream>>>(ws_imp);
    router_kernel<<<B / ROWS_PER_BLOCK, 128, 0, stream>>>(
        hidden, gate_w, out_idx, out_w, ws_imp, ws_load);
    aux_kernel<<<1, NUM_EXPERTS, 0, stream>>>(ws_imp, ws_load, out_aux, B);
}